// PMLP_GCN_80083960201233
// MI455X (gfx1250) — compile-verified
//
#include <hip/hip_runtime.h>

typedef __attribute__((ext_vector_type(2))) float v2f;
typedef __attribute__((ext_vector_type(8))) float v8f;
typedef __attribute__((ext_vector_type(4))) int   v4i;

#define HDIM 128
#define LDS_STRIDE 132   // pad 128->132 so B-operand LDS reads spread across banks

#if __has_builtin(__builtin_amdgcn_global_load_async_to_lds_b128)
#define HAVE_ASYNC_LDS 1
typedef __attribute__((address_space(1))) v4i* gptr_b128;
typedef __attribute__((address_space(3))) v4i* lptr_b128;
#else
#define HAVE_ASYNC_LDS 0
#endif

// ---------------------------------------------------------------- degree
__global__ void deg_kernel(const int* __restrict__ col, const float* __restrict__ w,
                           float* __restrict__ deg, int E) {
    int t = blockIdx.x * blockDim.x + threadIdx.x;
    if (t < E) unsafeAtomicAdd(&deg[col[t]], w[t]);
}

__global__ void dinv_kernel(const float* __restrict__ deg, float* __restrict__ dinv, int N) {
    int t = blockIdx.x * blockDim.x + threadIdx.x;
    if (t < N) {
        float d = deg[t];
        dinv[t] = d > 0.f ? rsqrtf(d) : 0.f;
    }
}

// ---------------------------------------------------------------- fp32 WMMA GEMM
// Out[N,128] = A[N,128] @ W[128,128]^T. FUSE_BN: a = relu((A - mu) * rstd).
// One wave computes a 16x128 output tile with V_WMMA_F32_16X16X4_F32.
template <bool FUSE_BN>
__global__ __launch_bounds__(256) void gemm_wmma_kernel(
    const float* __restrict__ A, const float* __restrict__ W,
    float* __restrict__ Out, const float* __restrict__ mu,
    const float* __restrict__ rstd, int N)
{
    __shared__ float sW[HDIM * LDS_STRIDE];           // 67.5 KB of the 320 KB WGP LDS
    const int tid = threadIdx.x;

#if HAVE_ASYNC_LDS
    // Async global->LDS copy engine (ASYNCcnt), b128 chunks, no VGPR staging.
    for (int i = tid; i < HDIM * 32; i += 256) {      // 4096 x 16B transfers
        int n = i >> 5, c = (i & 31) * 4;
        __builtin_amdgcn_global_load_async_to_lds_b128(
            (gptr_b128)(W + n * HDIM + c),
            (lptr_b128)(sW + n * LDS_STRIDE + c),
            0, 0);
    }
#if __has_builtin(__builtin_amdgcn_s_wait_asynccnt)
    __builtin_amdgcn_s_wait_asynccnt(0);
#else
    asm volatile("s_wait_asynccnt 0" ::: "memory");
#endif
#else
    for (int i = tid; i < HDIM * HDIM; i += 256) {    // fallback: synchronous stage
        int n = i >> 7, k = i & 127;
        sW[n * LDS_STRIDE + k] = W[i];
    }
#endif
    __syncthreads();

    const int wave  = tid >> 5;
    const int lane  = tid & 31;
    const int mloc  = lane & 15;          // A-row / B-col within 16-tile (ISA 7.12.2)
    const int khalf = (lane >> 4) * 2;    // lane-half selects K pair {0,1} or {2,3}
    const long row0 = (long)blockIdx.x * 128 + wave * 16;
    const int rowA  = (int)row0 + mloc;
    const int rowC  = rowA < N ? rowA : 0;            // clamp; garbage rows never stored
    const float* aRow = A + (long)rowC * HDIM;

    const v8f vzero = {0.f, 0.f, 0.f, 0.f, 0.f, 0.f, 0.f, 0.f};
    v8f acc[8];
#pragma unroll
    for (int i = 0; i < 8; ++i) acc[i] = vzero;

    for (int k0 = 0; k0 < HDIM; k0 += 4) {
        const int kk = k0 + khalf;
        float2 av = *(const float2*)(aRow + kk);      // A: M=mloc, K=kk..kk+1
        if constexpr (FUSE_BN) {                      // branch-free: compile-time
            av.x = fmaxf((av.x - mu[kk    ]) * rstd[kk    ], 0.f);
            av.y = fmaxf((av.y - mu[kk + 1]) * rstd[kk + 1], 0.f);
        }
        v2f a; a.x = av.x; a.y = av.y;
#pragma unroll
        for (int nt = 0; nt < 8; ++nt) {
            // B[k][n] = W[n][k] : lane holds n = nt*16+mloc, K=kk..kk+1 (LDS b64)
            const float* bp = &sW[(nt * 16 + mloc) * LDS_STRIDE + kk];
            v2f b; b.x = bp[0]; b.y = bp[1];
            acc[nt] = __builtin_amdgcn_wmma_f32_16x16x4_f32(
                false, a, false, b, (short)0, acc[nt], false, false);
        }
    }

    // D layout: VGPR j -> row = row0 + (lane<16 ? j : 8+j), col = nt*16 + mloc
    const int rOff = (lane < 16) ? 0 : 8;
#pragma unroll
    for (int nt = 0; nt < 8; ++nt) {
#pragma unroll
        for (int j = 0; j < 8; ++j) {
            long r = row0 + rOff + j;
            if (r < N) Out[r * HDIM + nt * 16 + mloc] = acc[nt][j];
        }
    }
}

// ---------------------------------------------------------------- edge scatter
// out[col] += h[row] * (dinv[row]*dinv[col]*w); 32 lanes per edge, 4 feats/lane.
__global__ void scatter_kernel(const float* __restrict__ h, const int* __restrict__ row,
                               const int* __restrict__ col, const float* __restrict__ w,
                               const float* __restrict__ dinv, float* __restrict__ out,
                               int E)
{
    long t = (long)blockIdx.x * blockDim.x + threadIdx.x;
    int e  = (int)(t >> 5);
    if (e >= E) return;
    int f4 = ((int)t & 31) * 4;
    int r = row[e], c = col[e];
    float norm = dinv[r] * dinv[c] * w[e];
    const float4 v = *(const float4*)(h + (long)r * HDIM + f4);
    float* dst = out + (long)c * HDIM + f4;
    unsafeAtomicAdd(dst + 0, v.x * norm);
    unsafeAtomicAdd(dst + 1, v.y * norm);
    unsafeAtomicAdd(dst + 2, v.z * norm);
    unsafeAtomicAdd(dst + 3, v.w * norm);
}

// ---------------------------------------------------------------- batchnorm stats
__global__ void stats_kernel(const float* __restrict__ hbuf, float* __restrict__ sums, int N) {
    int f = threadIdx.x & 127;
    int sub = threadIdx.x >> 7;              // 2 node-rows per block
    int n0 = blockIdx.x * 2 + sub;
    int ns = gridDim.x * 2;
    float s = 0.f, sq = 0.f;
    for (int n = n0; n < N; n += ns) {
        float v = hbuf[(long)n * HDIM + f];
        s += v; sq += v * v;
    }
    unsafeAtomicAdd(&sums[f], s);
    unsafeAtomicAdd(&sums[128 + f], sq);
}

__global__ void finalize_stats_kernel(const float* __restrict__ sums,
                                      float* __restrict__ mustd, int N) {
    int f = threadIdx.x;
    if (f < 128) {
        float inv  = 1.f / (float)N;
        float mean = sums[f] * inv;
        float var  = sums[128 + f] * inv - mean * mean;
        mustd[f]       = mean;
        mustd[128 + f] = rsqrtf(var + 1e-5f);
    }
}

// ---------------------------------------------------------------- out = b1 broadcast
__global__ void init_out_kernel(const float* __restrict__ b, float* __restrict__ out, long total) {
    long t = (long)blockIdx.x * blockDim.x + threadIdx.x;
    if (t < total) out[t] = b[t & 127];
}

// ================================================================ launch
extern "C" void kernel_launch(void* const* d_in, const int* in_sizes, int n_in,
                              void* d_out, int out_size, void* d_ws, size_t ws_size,
                              hipStream_t stream) {
    (void)n_in; (void)out_size; (void)ws_size;
    const float* x  = (const float*)d_in[0];
    const int*   ei = (const int*)d_in[1];
    const float* ew = (const float*)d_in[2];
    const float* W0 = (const float*)d_in[3];
    const float* b0 = (const float*)d_in[4]; (void)b0; // cancels inside BatchNorm
    const float* W1 = (const float*)d_in[5];
    const float* b1 = (const float*)d_in[6];
    float* out = (float*)d_out;

    const int N = in_sizes[0] / HDIM;
    const int E = in_sizes[2];
    const int* row = ei;
    const int* col = ei + E;

    char* ws = (char*)d_ws;
    size_t off = 0;
    auto carve = [&](size_t bytes) {
        char* p = ws + off;
        off = (off + bytes + 255) & ~(size_t)255;
        return p;
    };
    float* h0    = (float*)carve((size_t)N * HDIM * sizeof(float));
    float* h1    = (float*)carve((size_t)N * HDIM * sizeof(float));
    float* deg   = (float*)carve((size_t)N * sizeof(float));
    float* dinv  = (float*)carve((size_t)N * sizeof(float));
    float* sums  = (float*)carve(256 * sizeof(float));
    float* mustd = (float*)carve(256 * sizeof(float));

    (void)hipMemsetAsync(deg, 0, (size_t)N * sizeof(float), stream);
    (void)hipMemsetAsync(h1, 0, (size_t)N * HDIM * sizeof(float), stream);
    (void)hipMemsetAsync(sums, 0, 256 * sizeof(float), stream);

    deg_kernel<<<(E + 255) / 256, 256, 0, stream>>>(col, ew, deg, E);
    dinv_kernel<<<(N + 255) / 256, 256, 0, stream>>>(deg, dinv, N);

    const int gemmBlocks = (N + 127) / 128;
    gemm_wmma_kernel<false><<<gemmBlocks, 256, 0, stream>>>(x, W0, h0, nullptr, nullptr, N);

    const long st = (long)E * 32;
    scatter_kernel<<<(int)((st + 255) / 256), 256, 0, stream>>>(h0, row, col, ew, dinv, h1, E);

    stats_kernel<<<1024, 256, 0, stream>>>(h1, sums, N);
    finalize_stats_kernel<<<1, 128, 0, stream>>>(sums, mustd, N);

    // GEMM1 with fused BatchNorm+ReLU on the A operand (compile-time specialization)
    gemm_wmma_kernel<true><<<gemmBlocks, 256, 0, stream>>>(h1, W1, h0, mustd, mustd + 128, N);

    const long total = (long)N * HDIM;
    init_out_kernel<<<(int)((total + 255) / 256), 256, 0, stream>>>(b1, out, total);
    scatter_kernel<<<(int)((st + 255) / 256), 256, 0, stream>>>(h0, row, col, ew, dinv, out, E);
}